// vis_aware_track_cvae_33621003993145
// MI455X (gfx1250) — compile-verified
//
#include <hip/hip_runtime.h>

#define NVARC 11
#define NUMT  100
#define MAXIT 50
#define NB    2048
#define NCHUNK 7
#define TPB   (NCHUNK * 32)   // one wave per time chunk

typedef float v2f __attribute__((ext_vector_type(2)));
typedef float v8f __attribute__((ext_vector_type(8)));

static __device__ __forceinline__ v8f wmma4(v2f a, v2f b, v8f c) {
  // D = A(16x4 f32) * B(4x16 f32) + C(16x16 f32)
  return __builtin_amdgcn_wmma_f32_16x16x4_f32(false, a, false, b, (short)0, c, false, false);
}

static __device__ __forceinline__ float getrow(const v8f& v, int i) {
  switch (i & 7) {
    case 0: return v[0]; case 1: return v[1]; case 2: return v[2]; case 3: return v[3];
    case 4: return v[4]; case 5: return v[5]; case 6: return v[6]; default: return v[7];
  }
}

// Full D-layout -> B-operand element (used for the small nvar-indexed conversions)
static __device__ __forceinline__ float hmerge(const v8f& d, int lo, int hi, int hlf) {
  float vlo = (lo < 8) ? getrow(d, lo) : __shfl_xor(getrow(d, lo - 8), 16, 32);
  float vhi = (hi >= 8) ? getrow(d, hi - 8) : __shfl_xor(getrow(d, hi), 16, 32);
  return hlf ? vhi : vlo;
}

// 4-row half-pass fragment builder: rows d0..d3 are local rows R..R+3 of each
// lane half.  lowg (g = hp): lower-half times; highg (g = hp+2): upper-half times.
static __device__ __forceinline__ v2f bfrag(float d0, float d1, float d2, float d3,
                                            int hlf, bool highg) {
  v2f r;
  if (highg) {
    float s0 = __shfl_xor(d0, 16, 32), s1 = __shfl_xor(d1, 16, 32);
    r.x = hlf ? d2 : s0;
    r.y = hlf ? d3 : s1;
  } else {
    float s2 = __shfl_xor(d2, 16, 32), s3 = __shfl_xor(d3, 16, 32);
    r.x = hlf ? s2 : d0;
    r.y = hlf ? s3 : d1;
  }
  return r;
}

// ---------------- Precompute: KKT inverses via Gauss-Jordan ----------------

static __device__ void gaussjordan(float* A, int N, int W) {
#pragma unroll 1
  for (int col = 0; col < N; ++col) {
    int p = col; float best = fabsf(A[col * W + col]);
#pragma unroll 1
    for (int r = col + 1; r < N; ++r) {
      float v = fabsf(A[r * W + col]);
      if (v > best) { best = v; p = r; }
    }
    if (p != col) {
#pragma unroll 1
      for (int c = 0; c < W; ++c) { float t = A[col * W + c]; A[col * W + c] = A[p * W + c]; A[p * W + c] = t; }
    }
    float ip = 1.f / A[col * W + col];
#pragma unroll 1
    for (int c = 0; c < W; ++c) A[col * W + c] *= ip;
#pragma unroll 1
    for (int r = 0; r < N; ++r) {
      if (r == col) continue;
      float f = A[r * W + col];
      if (f != 0.f)
#pragma unroll 1
        for (int c = 0; c < W; ++c) A[r * W + c] = fmaf(-f, A[col * W + c], A[r * W + c]);
    }
  }
}

// ws layout (floats): [0,289) inv_init 17x17 | [289,514) inv_opt 15x15 | [514,635) S 11x11
__global__ void precompute_kernel(const float* __restrict__ P, const float* __restrict__ Pd,
                                  const float* __restrict__ Pdd, float* __restrict__ ws) {
  __shared__ float gP[121], gV[121], gA[121];
  __shared__ float Aug[17 * 34];
  int tid = threadIdx.x;
#pragma unroll 1
  for (int idx = tid; idx < 121; idx += blockDim.x) {
    int j = idx / 11, k = idx % 11;
    float sp = 0.f, sv = 0.f, sa = 0.f;
#pragma unroll 1
    for (int t = 0; t < NUMT; ++t) {
      sp = fmaf(P[t * 11 + j],  P[t * 11 + k],  sp);
      sv = fmaf(Pd[t * 11 + j], Pd[t * 11 + k], sv);
      sa = fmaf(Pdd[t * 11 + j], Pdd[t * 11 + k], sa);
    }
    gP[idx] = sp; gV[idx] = sv; gA[idx] = sa;
    // S = cost_opt - I = 11*P^TP + Pdot^TPdot + Pddot^TPddot
    ws[514 + idx] = 11.f * sp + sv + sa;
  }
  __syncthreads();
  if (tid == 0) {
    // M_init (17x17): [Pddot^T Pddot, Aeq_init^T; Aeq_init, 0]
    const float* eq6[6] = { P, Pd, P + 99 * 11, Pd + 99 * 11, P + 49 * 11, P + 74 * 11 };
    const int N1 = 17, W1 = 34;
    for (int r = 0; r < N1; ++r) for (int c = 0; c < W1; ++c) Aug[r * W1 + c] = 0.f;
    for (int r = 0; r < 11; ++r) for (int c = 0; c < 11; ++c) Aug[r * W1 + c] = gA[r * 11 + c];
    for (int e = 0; e < 6; ++e)
      for (int k = 0; k < 11; ++k) { Aug[k * W1 + 11 + e] = eq6[e][k]; Aug[(11 + e) * W1 + k] = eq6[e][k]; }
    for (int r = 0; r < N1; ++r) Aug[r * W1 + N1 + r] = 1.f;
    gaussjordan(Aug, N1, W1);
    for (int r = 0; r < N1; ++r) for (int c = 0; c < N1; ++c) ws[r * 17 + c] = Aug[r * W1 + N1 + c];
  }
  __syncthreads();
  if (tid == 0) {
    // M_opt (15x15): cost = S + I
    const float* eq4[4] = { P, Pd, P + 99 * 11, Pd + 99 * 11 };
    const int N2 = 15, W2 = 30;
    for (int r = 0; r < N2; ++r) for (int c = 0; c < W2; ++c) Aug[r * W2 + c] = 0.f;
    for (int r = 0; r < 11; ++r)
      for (int c = 0; c < 11; ++c)
        Aug[r * W2 + c] = ws[514 + r * 11 + c] + (r == c ? 1.f : 0.f);
    for (int e = 0; e < 4; ++e)
      for (int k = 0; k < 11; ++k) { Aug[k * W2 + 11 + e] = eq4[e][k]; Aug[(11 + e) * W2 + k] = eq4[e][k]; }
    for (int r = 0; r < N2; ++r) Aug[r * W2 + N2 + r] = 1.f;
    gaussjordan(Aug, N2, W2);
    for (int r = 0; r < N2; ++r) for (int c = 0; c < N2; ++c) ws[289 + r * 15 + c] = Aug[r * W2 + N2 + c];
  }
}

// ---------- Main solver: one block per 16-batch tile, one wave per chunk ----------

__global__ __launch_bounds__(TPB, 1) void solver_kernel(
    const float* __restrict__ gPm, const float* __restrict__ gPdm, const float* __restrict__ gPddm,
    const float* __restrict__ ego, const float* __restrict__ term,
    const float* __restrict__ via, const float* __restrict__ obs,
    const float* __restrict__ tgt, const float* __restrict__ bnd,
    const float* __restrict__ ws, float* __restrict__ out) {
  __shared__ __align__(16) float sP[3][112][12];   // padded basis, row-major [time][nvar]
  __shared__ __align__(16) float sPT[3][16][112];  // padded basis, transposed [nvar][time]
  __shared__ __align__(16) float Qbuf[NCHUNK][32][20];  // partial Q1 exchange (padded rows)
  __shared__ float sG[16][12];                     // inv_opt[0:11,0:11] padded
  __shared__ float sS[16][12];                     // S = cost_opt - I, padded
  __shared__ float sKi[16][6];                     // inv_init[j][11+m]
  __shared__ float sKo[16][4];                     // inv_opt[j][11+m]

  const float* mats[3] = { gPm, gPdm, gPddm };
#pragma unroll 1
  for (int idx = threadIdx.x; idx < 3 * 112 * 12; idx += TPB) {
    int m = idx / (112 * 12); int rem = idx % (112 * 12);
    int t = rem / 12, k = rem % 12;
    sP[m][t][k] = (t < NUMT && k < NVARC) ? mats[m][t * NVARC + k] : 0.f;
  }
#pragma unroll 1
  for (int idx = threadIdx.x; idx < 3 * 16 * 112; idx += TPB) {
    int m = idx / (16 * 112); int rem = idx % (16 * 112);
    int k = rem / 112, t = rem % 112;
    sPT[m][k][t] = (t < NUMT && k < NVARC) ? mats[m][t * NVARC + k] : 0.f;
  }
  const float* inv_i = ws;
  const float* inv_o = ws + 289;
  const float* Smat  = ws + 514;
#pragma unroll 1
  for (int idx = threadIdx.x; idx < 16 * 12; idx += TPB) {
    int j = idx / 12, k = idx % 12;
    sG[j][k] = (j < NVARC && k < NVARC) ? inv_o[j * 15 + k] : 0.f;
    sS[j][k] = (j < NVARC && k < NVARC) ? Smat[j * 11 + k] : 0.f;
  }
#pragma unroll 1
  for (int idx = threadIdx.x; idx < 16 * 6; idx += TPB) {
    int j = idx / 6, m = idx % 6;
    sKi[j][m] = (j < NVARC) ? inv_i[j * 17 + 11 + m] : 0.f;
  }
#pragma unroll 1
  for (int idx = threadIdx.x; idx < 16 * 4; idx += TPB) {
    int j = idx / 4, m = idx % 4;
    sKo[j][m] = (j < NVARC) ? inv_o[j * 15 + 11 + m] : 0.f;
  }
  __syncthreads();

  const int lane = threadIdx.x & 31;
  const int wid  = threadIdx.x >> 5;  // wave id == time-chunk id (0..6)
  const int hlf  = lane >> 4;         // 0: rows 0-7, 1: rows 8-15 of D fragments
  const int coln = lane & 15;         // batch within tile / matrix row index
  const int b    = blockIdx.x * 16 + coln;

  // per-batch scalars (duplicated across lane halves; same tile for all waves)
  const float vx0 = ego[b * 4 + 2], vy0 = ego[b * 4 + 3];
  const float xtm = term[b * 4 + 0], ytm = term[b * 4 + 1];
  const float vxt = term[b * 4 + 2], vyt = term[b * 4 + 3];
  const float vp0 = via[b * 4 + 0], vp1 = via[b * 4 + 1];
  const float vp2 = via[b * 4 + 2], vp3 = via[b * 4 + 3];
  float xo[10], yo[10];
#pragma unroll
  for (int o = 0; o < 10; ++o) { xo[o] = obs[b * 20 + o]; yo[o] = obs[b * 20 + 10 + o]; }
  const float tx0 = tgt[b * 4 + 0], ty0 = tgt[b * 4 + 1];
  const float tvx = tgt[b * 4 + 2], tvy = tgt[b * 4 + 3];
  const float dmn = bnd[b * 2 + 0], dmx = bnd[b * 2 + 1];

  const v8f zv = { 0.f, 0.f, 0.f, 0.f, 0.f, 0.f, 0.f, 0.f };

  // c_x_bar / c_y_bar and constant part of the per-iteration solve, in D layout
  v8f cbx = zv, cby = zv, ccx = zv, ccy = zv;
#pragma unroll
  for (int r = 0; r < 8; ++r) {
    int j = r + 8 * hlf;
    cbx[r] = vx0 * sKi[j][1] + xtm * sKi[j][2] + vxt * sKi[j][3] + vp0 * sKi[j][4] + vp1 * sKi[j][5];
    cby[r] = vy0 * sKi[j][1] + ytm * sKi[j][2] + vyt * sKi[j][3] + vp2 * sKi[j][4] + vp3 * sKi[j][5];
    ccx[r] = vx0 * sKo[j][1] + xtm * sKo[j][2] + vxt * sKo[j][3];
    ccy[r] = vy0 * sKo[j][1] + ytm * sKo[j][2] + vyt * sKo[j][3];
  }

  // redundant per-wave state (identical arithmetic in every wave keeps it consistent)
  v8f cxD = cbx, cyD = cby;   // primal coefficients, D layout [nvar][batch]
  v8f lxD = zv,  lyD = zv;    // lambdas, D layout

  const int c = wid;          // this wave's time chunk

#pragma unroll 1
  for (int it = 0; it < MAXIT; ++it) {
    // coefficients -> B-operand layout (B[k=nvar][n=batch])
    float cxB[6], cyB[6];
#pragma unroll
    for (int s = 0; s < 3; ++s) {
      cxB[2 * s + 0] = hmerge(cxD, 4 * s + 0, 4 * s + 2, hlf);
      cxB[2 * s + 1] = hmerge(cxD, 4 * s + 1, 4 * s + 3, hlf);
      cyB[2 * s + 0] = hmerge(cyD, 4 * s + 0, 4 * s + 2, hlf);
      cyB[2 * s + 1] = hmerge(cyD, 4 * s + 1, 4 * s + 3, hlf);
    }

    // ---- forward: trajectories for this wave's 16-sample time chunk
    v8f X = zv, XD = zv, XDD = zv, Y = zv, YD = zv, YDD = zv;
    const int trow = c * 16 + coln;
#pragma unroll
    for (int s = 0; s < 3; ++s) {
      const int cc = 4 * s + 2 * hlf;
      v2f aP  = { sP[0][trow][cc], sP[0][trow][cc + 1] };
      v2f aV  = { sP[1][trow][cc], sP[1][trow][cc + 1] };
      v2f aA  = { sP[2][trow][cc], sP[2][trow][cc + 1] };
      v2f bx = { cxB[2 * s], cxB[2 * s + 1] };
      v2f by = { cyB[2 * s], cyB[2 * s + 1] };
      X   = wmma4(aP, bx, X);   Y   = wmma4(aP, by, Y);
      XD  = wmma4(aV, bx, XD);  YD  = wmma4(aV, by, YD);
      XDD = wmma4(aA, bx, XDD); YDD = wmma4(aA, by, YDD);
    }

    // ---- elementwise + partial transposed reductions (two 4-row half-passes)
    v8f Q1xp = zv, Q1yp = zv;
#pragma unroll
    for (int hp = 0; hp < 2; ++hp) {
      float p1x[4], d1x[4], a1x[4], p1y[4], d1y[4], a1y[4];
#pragma unroll
      for (int r = 0; r < 4; ++r) {
        const int R = hp * 4 + r;
        const int t = c * 16 + R + 8 * hlf;
        const float x = X[R], y = Y[R];
        const float xd = XD[R], yd = YD[R];
        const float xdd = XDD[R], ydd = YDD[R];

        // obstacles: cos/sin(atan2) via rsqrt; A_OBS = B_OBS = 0.4, c1 == 0.16
        float rsx = 0.f, rsy = 0.f;
#pragma unroll
        for (int o = 0; o < 10; ++o) {
          float wc = x - xo[o], wv = y - yo[o];
          float pp = wc * 0.4f, qq = wv * 0.4f;
          float h2 = pp * pp + qq * qq;
          float rin = (h2 > 0.f) ? rsqrtf(h2) : 0.f;
          float ca = (h2 > 0.f) ? pp * rin : 1.f;
          float sa = qq * rin;
          float c2v = 0.4f * (wc * ca + wv * sa);
          float dd = fmaxf(1.f, c2v * 6.25f);   // /0.16
          rsx += wc - 0.4f * dd * ca;
          rsy += wv - 0.4f * dd * sa;
        }
        // velocity clip
        float v2 = xd * xd + yd * yd;
        float vin = (v2 > 0.f) ? rsqrtf(v2) : 0.f;
        float cv = (v2 > 0.f) ? xd * vin : 1.f;
        float sv = yd * vin;
        float dv = fminf(fmaxf(v2 * vin, 0.1f), 2.0f);
        float rvx = xd - dv * cv, rvy = yd - dv * sv;
        // acceleration clip
        float a2 = xdd * xdd + ydd * ydd;
        float ain = (a2 > 0.f) ? rsqrtf(a2) : 0.f;
        float cA = (a2 > 0.f) ? xdd * ain : 1.f;
        float sA = ydd * ain;
        float da = fminf(a2 * ain, 2.0f);
        float rax = xdd - da * cA, ray = ydd - da * sA;
        // target band
        float tt = (float)t * (5.0f / 99.0f);
        float wct = x - (tx0 + tvx * tt);
        float wst = y - (ty0 + tvy * tt);
        float t2 = wct * wct + wst * wst;
        float tin = (t2 > 0.f) ? rsqrtf(t2) : 0.f;
        float cb = (t2 > 0.f) ? wct * tin : 1.f;
        float sb = wst * tin;
        float dtv = fminf(fmaxf(t2 * tin, dmn), dmx);
        float rtx = wct - dtv * cb, rty = wst - dtv * sb;

        float msk = (t < NUMT) ? 1.f : 0.f;
        p1x[r] = (rsx + rtx) * msk; d1x[r] = rvx * msk; a1x[r] = rax * msk;
        p1y[r] = (rsy + rty) * msk; d1y[r] = rvy * msk; a1y[r] = ray * msk;
      }

      // transposed reductions for this half-pass (g = hp and g = hp+2)
#pragma unroll
      for (int gi = 0; gi < 2; ++gi) {
        const int g = hp + 2 * gi;
        const bool highg = (gi == 1);
        const int tc = c * 16 + 4 * g + 2 * hlf;
        v2f aPT = { sPT[0][coln][tc], sPT[0][coln][tc + 1] };
        v2f aVT = { sPT[1][coln][tc], sPT[1][coln][tc + 1] };
        v2f aAT = { sPT[2][coln][tc], sPT[2][coln][tc + 1] };
        v2f bp1x = bfrag(p1x[0], p1x[1], p1x[2], p1x[3], hlf, highg);
        v2f bd1x = bfrag(d1x[0], d1x[1], d1x[2], d1x[3], hlf, highg);
        v2f ba1x = bfrag(a1x[0], a1x[1], a1x[2], a1x[3], hlf, highg);
        v2f bp1y = bfrag(p1y[0], p1y[1], p1y[2], p1y[3], hlf, highg);
        v2f bd1y = bfrag(d1y[0], d1y[1], d1y[2], d1y[3], hlf, highg);
        v2f ba1y = bfrag(a1y[0], a1y[1], a1y[2], a1y[3], hlf, highg);
        Q1xp = wmma4(aPT, bp1x, Q1xp); Q1xp = wmma4(aVT, bd1x, Q1xp); Q1xp = wmma4(aAT, ba1x, Q1xp);
        Q1yp = wmma4(aPT, bp1y, Q1yp); Q1yp = wmma4(aVT, bd1y, Q1yp); Q1yp = wmma4(aAT, ba1y, Q1yp);
      }
    }

    // F = (cost_opt - I) @ c  (small, recomputed redundantly per wave)
    v8f Fx = zv, Fy = zv;
#pragma unroll
    for (int s = 0; s < 3; ++s) {
      const int cc = 4 * s + 2 * hlf;
      v2f aS = { sS[coln][cc], sS[coln][cc + 1] };
      v2f bx = { cxB[2 * s], cxB[2 * s + 1] };
      v2f by = { cyB[2 * s], cyB[2 * s + 1] };
      Fx = wmma4(aS, bx, Fx);
      Fy = wmma4(aS, by, Fy);
    }

    // ---- cross-wave reduction of Q1 partials through LDS
    *(v8f*)&Qbuf[wid][lane][0] = Q1xp;
    *(v8f*)&Qbuf[wid][lane][8] = Q1yp;
    __syncthreads();
    v8f Q1x = zv, Q1y = zv;
#pragma unroll
    for (int w = 0; w < NCHUNK; ++w) {
      Q1x += *(const v8f*)&Qbuf[w][lane][0];
      Q1y += *(const v8f*)&Qbuf[w][lane][8];
    }
    __syncthreads();  // protect Qbuf before next iteration's writes

    // lambda update, then -lin = lx_new + (F - Q1) + c_bar   (Q2 = F - Q1 identity)
    lxD = lxD - Q1x;
    lyD = lyD - Q1y;
    v8f nlx = lxD + Fx - Q1x + cbx;
    v8f nly = lyD + Fy - Q1y + cby;

    // new coefficients: c = G @ (-lin) + const_part (redundant per wave)
    v8f dcx = ccx, dcy = ccy;
#pragma unroll
    for (int s = 0; s < 3; ++s) {
      const int cc = 4 * s + 2 * hlf;
      v2f aG = { sG[coln][cc], sG[coln][cc + 1] };
      v2f bnx = { hmerge(nlx, 4 * s, 4 * s + 2, hlf), hmerge(nlx, 4 * s + 1, 4 * s + 3, hlf) };
      v2f bny = { hmerge(nly, 4 * s, 4 * s + 2, hlf), hmerge(nly, 4 * s + 1, 4 * s + 3, hlf) };
      dcx = wmma4(aG, bnx, dcx);
      dcy = wmma4(aG, bny, dcy);
    }
    cxD = dcx; cyD = dcy;
  }

  // write primal = [cx(11) | cy(11)] per batch (one wave only)
  if (wid == 0) {
#pragma unroll
    for (int r = 0; r < 8; ++r) {
      int j = r + 8 * hlf;
      if (j < NVARC) {
        out[b * 22 + j] = cxD[r];
        out[b * 22 + 11 + j] = cyD[r];
      }
    }
  }
}

extern "C" void kernel_launch(void* const* d_in, const int* in_sizes, int n_in,
                              void* d_out, int out_size, void* d_ws, size_t ws_size,
                              hipStream_t stream) {
  (void)in_sizes; (void)n_in; (void)out_size; (void)ws_size;
  const float* P    = (const float*)d_in[0];
  const float* Pd   = (const float*)d_in[1];
  const float* Pdd  = (const float*)d_in[2];
  const float* ego  = (const float*)d_in[3];
  const float* term = (const float*)d_in[4];
  const float* via  = (const float*)d_in[5];
  const float* obs  = (const float*)d_in[6];
  const float* tgt  = (const float*)d_in[7];
  const float* bnd  = (const float*)d_in[8];
  float* ws  = (float*)d_ws;
  float* out = (float*)d_out;

  precompute_kernel<<<dim3(1), dim3(128), 0, stream>>>(P, Pd, Pdd, ws);
  solver_kernel<<<dim3(NB / 16), dim3(TPB), 0, stream>>>(
      P, Pd, Pdd, ego, term, via, obs, tgt, bnd, ws, out);
}